// KAN2_30588757082464
// MI455X (gfx1250) — compile-verified
//
#include <hip/hip_runtime.h>
#include <cstdint>

// KAN2: B=1M samples, Q=5 branches, D=3 inputs, H=16 hidden, complex64 out.
// Trans-bound (~320 silu/sample); memory floor ~0.9us @ 23.3 TB/s.
// Strategy: wave32, 4 samples/lane (ILP for the quarter-rate trans pipe,
// amortizes LDS issue 4x), psi+phi fused per branch q with a dynamic q loop
// (no spills), silu via ONE hardware trans op: z/2 + (z/2)*v_tanh_f32(z/2),
// weights (4KB) async-staged to LDS once per block (CDNA5 ASYNCcnt path),
// NT hints on the streaming x/t/out traffic.

#define BQ 5
#define BD 3
#define BH 16
#define SPT 4           // samples per thread
#define TPB 256         // threads per block

typedef float v2f __attribute__((ext_vector_type(2)));  // true vector type
                                                        // (NT builtins reject
                                                        //  HIP_vector_type)

// LDS layout (float offsets)
#define OW1   0     // 240
#define OB1   240   // 240
#define OW2   480   // 240
#define OB2   720   // 15
#define OPW1  736   // 80  (16B aligned)
#define OPB1  816   // 80
#define OPW2  896   // 80
#define OPB2  976   // 5
#define OWF   981   // 10
#define OBF   991   // 2
#define SM_FLOATS 996

__device__ __forceinline__ float fast_tanh(float v) {
#if __has_builtin(__builtin_amdgcn_tanhf)
  return __builtin_amdgcn_tanhf(v);
#else
  float r;
  asm("v_tanh_f32 %0, %1" : "=v"(r) : "v"(v));
  return r;
#endif
}

__device__ __forceinline__ float fast_silu(float z) {
  // silu(z) = z*sigmoid(z) = t + t*tanh(t), t = z/2.
  // 1 trans (v_tanh_f32) + 2 valu, vs 2 trans + 3 valu for exp/rcp form.
  const float t = 0.5f * z;
  return fmaf(t, fast_tanh(t), t);
}

__global__ __launch_bounds__(TPB) void kan2_kernel(
    const float* __restrict__ x,   const float* __restrict__ t,
    const float* __restrict__ w1,  const float* __restrict__ b1,
    const float* __restrict__ w2,  const float* __restrict__ b2,
    const float* __restrict__ pw1, const float* __restrict__ pb1,
    const float* __restrict__ pw2, const float* __restrict__ pb2,
    const float* __restrict__ wf,  const float* __restrict__ bf,
    float* __restrict__ out, int n) {
  __shared__ __align__(16) float sm[SM_FLOATS];
  const unsigned lds_base = (unsigned)(uintptr_t)(&sm[0]);
  const int tid = threadIdx.x;

  // --- Stage all weights into LDS via CDNA5 async global->LDS copies ------
  auto stage = [&](const float* src, int count, int off) {
    for (int k = tid; k < count; k += TPB) {
      unsigned la = lds_base + (unsigned)((off + k) << 2); // LDS byte addr
      unsigned ga = (unsigned)(k << 2);                    // byte off vs saddr
      asm volatile("global_load_async_to_lds_b32 %0, %1, %2 offset:0"
                   :: "v"(la), "v"(ga), "s"(src) : "memory");
    }
  };
  stage(w1, 240, OW1);  stage(b1, 240, OB1);  stage(w2, 240, OW2);
  stage(b2, 15, OB2);   stage(pw1, 80, OPW1); stage(pb1, 80, OPB1);
  stage(pw2, 80, OPW2); stage(pb2, 5, OPB2);  stage(wf, 10, OWF);
  stage(bf, 2, OBF);
  asm volatile("s_wait_asynccnt 0x0" ::: "memory");
  __syncthreads();

  // --- Four samples per lane (independent trans chains, shared weights) ---
  const int i0 = blockIdx.x * (TPB * SPT) + tid;
  int   idx[SPT];
  bool  valid[SPT];
#pragma unroll
  for (int s = 0; s < SPT; ++s) {
    const int i = i0 + s * TPB;
    valid[s] = (i < n);
    idx[s]   = valid[s] ? i : 0;
  }

  float xt[SPT][BD];
#pragma unroll
  for (int s = 0; s < SPT; ++s) {
    const v2f   xv = __builtin_nontemporal_load(&((const v2f*)x)[idx[s]]);
    const float tv = __builtin_nontemporal_load(&t[idx[s]]);
    xt[s][0] = xv.x; xt[s][1] = xv.y; xt[s][2] = tv;
  }

  float re[SPT], im[SPT];
  {
    const float bf0 = sm[OBF + 0], bf1 = sm[OBF + 1];
#pragma unroll
    for (int s = 0; s < SPT; ++s) { re[s] = bf0; im[s] = bf1; }
  }

  // --- Fused per-branch loop: psi (sum over d) then phi, accum into out ---
  // Dynamic q loop keeps static code & register pressure low (no spills);
  // d/h stay fully unrolled so xt[][] indexing is register-resident and LDS
  // reads vectorize to ds_load_b128.
#pragma unroll 1
  for (int q = 0; q < BQ; ++q) {
    float acc[SPT];
#pragma unroll
    for (int s = 0; s < SPT; ++s) acc[s] = 0.0f;

#pragma unroll
    for (int d = 0; d < BD; ++d) {
      const int base = (q * BD + d) * BH;
      float dv[SPT];
#pragma unroll
      for (int s = 0; s < SPT; ++s) dv[s] = 0.0f;
#pragma unroll
      for (int h = 0; h < BH; ++h) {
        const float w1v = sm[OW1 + base + h];   // one LDS read,
        const float b1v = sm[OB1 + base + h];   // shared by all
        const float w2v = sm[OW2 + base + h];   // four samples
#pragma unroll
        for (int s = 0; s < SPT; ++s) {
          const float z = fmaf(xt[s][d], w1v, b1v);
          dv[s] = fmaf(fast_silu(z), w2v, dv[s]);
        }
      }
      const float b2v = sm[OB2 + q * BD + d];
#pragma unroll
      for (int s = 0; s < SPT; ++s) acc[s] += dv[s] + b2v;
    }

    // phi[q] on s[q] = acc, immediately folded into the final projection
    float pv[SPT];
#pragma unroll
    for (int s = 0; s < SPT; ++s) pv[s] = 0.0f;
#pragma unroll
    for (int h = 0; h < BH; ++h) {
      const float pw1v = sm[OPW1 + q * BH + h];
      const float pb1v = sm[OPB1 + q * BH + h];
      const float pw2v = sm[OPW2 + q * BH + h];
#pragma unroll
      for (int s = 0; s < SPT; ++s) {
        const float z = fmaf(acc[s], pw1v, pb1v);
        pv[s] = fmaf(fast_silu(z), pw2v, pv[s]);
      }
    }
    const float pb2v = sm[OPB2 + q];
    const float wfr  = sm[OWF + q];       // wf[0, q]
    const float wfi  = sm[OWF + BQ + q];  // wf[1, q]
#pragma unroll
    for (int s = 0; s < SPT; ++s) {
      const float p = pv[s] + pb2v;
      re[s] = fmaf(p, wfr, re[s]);
      im[s] = fmaf(p, wfi, im[s]);
    }
  }

#pragma unroll
  for (int s = 0; s < SPT; ++s) {
    if (valid[s]) {
      v2f o; o.x = re[s]; o.y = im[s];
      __builtin_nontemporal_store(o, &((v2f*)out)[idx[s]]);
    }
  }
}

extern "C" void kernel_launch(void* const* d_in, const int* in_sizes, int n_in,
                              void* d_out, int out_size, void* d_ws, size_t ws_size,
                              hipStream_t stream) {
  (void)n_in; (void)d_ws; (void)ws_size; (void)out_size;
  const float* x   = (const float*)d_in[0];
  const float* t   = (const float*)d_in[1];
  const float* w1  = (const float*)d_in[2];
  const float* b1  = (const float*)d_in[3];
  const float* w2  = (const float*)d_in[4];
  const float* b2  = (const float*)d_in[5];
  const float* pw1 = (const float*)d_in[6];
  const float* pb1 = (const float*)d_in[7];
  const float* pw2 = (const float*)d_in[8];
  const float* pb2 = (const float*)d_in[9];
  const float* wf  = (const float*)d_in[10];
  const float* bf  = (const float*)d_in[11];
  float* out = (float*)d_out;
  const int n = in_sizes[1];  // B (element count of t)
  const int per_block = TPB * SPT;
  const int blocks = (n + per_block - 1) / per_block;
  kan2_kernel<<<blocks, TPB, 0, stream>>>(x, t, w1, b1, w2, b2,
                                          pw1, pb1, pw2, pb2, wf, bf, out, n);
}